// LearnableOrderGNN_28028956573740
// MI455X (gfx1250) — compile-verified
//
#include <hip/hip_runtime.h>
#include <cmath>
#include <cstring>
#include <cstdint>
#include <vector>
#include <algorithm>

typedef __attribute__((ext_vector_type(2))) float v2f;
typedef __attribute__((ext_vector_type(8))) float v8f;
typedef __attribute__((ext_vector_type(4))) unsigned int v4u;
typedef __attribute__((ext_vector_type(4))) int v4i;
typedef __attribute__((ext_vector_type(8))) int v8i;

static constexpr int    BB    = 128;    // batch of graphs
static constexpr int    NN    = 1024;   // nodes / signal length
static constexpr int    EEn   = 8192;   // edges per graph
static constexpr int    HIDc  = 128;    // hidden width
static constexpr int    NCLSc = 10;     // classes
#define HPconst 1.57079632679489661923f

// ---------------- device heap (single BSS symbol, ~276 MB) ----------------
static constexpr size_t OFF_E    = 0;                                  // E  [time][eig]
static constexpr size_t OFF_ET   = OFF_E   + (size_t)NN*NN;            // E^T [eig][time]
static constexpr size_t OFF_IDX  = OFF_ET  + (size_t)NN*NN;
static constexpr size_t OFF_GF   = OFF_IDX + NN;
static constexpr size_t OFF_BA   = OFF_GF  + NN;                       // 256x1024 ping
static constexpr size_t OFF_BBUF = OFF_BA  + (size_t)2*BB*NN;          // 256x1024 pong
static constexpr size_t OFF_Y    = OFF_BBUF+ (size_t)2*BB*NN;
static constexpr size_t OFF_W    = OFF_Y   + (size_t)BB*NN;
static constexpr size_t OFF_WN   = OFF_W   + (size_t)BB*EEn;
static constexpr size_t OFF_DEG  = OFF_WN  + (size_t)BB*EEn;
static constexpr size_t OFF_T1S  = OFF_DEG + (size_t)BB*NN;
static constexpr size_t OFF_T2S  = OFF_T1S + (size_t)BB*NN;
static constexpr size_t OFF_HA   = OFF_T2S + (size_t)BB*NN;
static constexpr size_t HSZ      = (size_t)BB*NN*HIDc;
static constexpr size_t OFF_HB   = OFF_HA + HSZ;
static constexpr size_t OFF_HC   = OFF_HB + HSZ;
static constexpr size_t OFF_HD   = OFF_HC + HSZ;
static constexpr size_t HEAP_TOTAL = OFF_HD + HSZ;

__device__ float g_heap[HEAP_TOTAL];

// ---------------- TDM 2-D tile load (Tensor Data Mover) -------------------
// Packs a D# per ISA ch.8: group0 {count=1, lds_addr, global_addr, type=2},
// group1 {data_size=4B, tensor dims == tile dims, tile WxH, row stride}.
// 2-D tensor -> groups 2/3 zero.  Completion tracked by TENSORcnt.
__device__ __forceinline__ void tdm_load_2d(unsigned ldsAddr, const void* gaddr,
                                            int tileW, int tileH, int strideElems)
{
    unsigned long long ga = (unsigned long long)(uintptr_t)gaddr;
    v4u g0;
    g0.x = 1u;                                              // count=1 (user D#)
    g0.y = ldsAddr;                                         // lds_addr (bytes)
    g0.z = (unsigned)(ga & 0xFFFFFFFFu);                    // global_addr[31:0]
    g0.w = (unsigned)((ga >> 32) & 0x1FFFFFFu) | (2u << 30);// ga[56:32] | type=2
    v8i g1;
    g1[0] = (int)(2u << 16);                                // data_size = 4 bytes
    g1[1] = (int)((unsigned)(tileW & 0xFFFF) << 16);        // tensor_dim0[15:0]
    g1[2] = (int)((unsigned)(tileH & 0xFFFF) << 16);        // tensor_dim1[15:0]
    g1[3] = (int)((unsigned)(tileW & 0xFFFF) << 16);        // tile_dim0
    g1[4] = (int)((unsigned)(tileH & 0xFFFF));              // tile_dim1 (tile_dim2=0)
    g1[5] = (int)((unsigned)strideElems);                   // tensor_dim0_stride[31:0]
    g1[6] = 0;                                              // stride0 hi / stride1 lo
    g1[7] = 0;
    v4i z4 = {0, 0, 0, 0};
#if __clang_major__ >= 23
    v8i z8 = {0, 0, 0, 0, 0, 0, 0, 0};
    __builtin_amdgcn_tensor_load_to_lds(g0, g1, z4, z4, z8, 0);
#else
    __builtin_amdgcn_tensor_load_to_lds(g0, g1, z4, z4, 0);
#endif
}

// ---------------- direct-global f32 WMMA GEMM (kept for M=128 FRFT) -------
__global__ __launch_bounds__(128) void gemm_f32_wmma(
    const float* __restrict__ A, const float* __restrict__ B,
    float* __restrict__ C, const float* __restrict__ bias,
    int M, int Nn, int K, int flags)
{
    const int lane = threadIdx.x & 31;
    const int wave = threadIdx.x >> 5;
    const int h    = lane >> 4;
    const int lm   = lane & 15;
    const int tilesX = Nn >> 4;
    const int totalTiles = (M >> 4) * tilesX;
    int tile = blockIdx.x * 4 + wave;
    if (tile >= totalTiles) return;
    const int tm = (tile / tilesX) << 4;
    const int tn = (tile % tilesX) << 4;

    const float* Ap = A + (size_t)(tm + lm) * K + 2 * h;
    const float* Bp = B + (size_t)(2 * h) * Nn + (tn + lm);
    __builtin_prefetch(Ap, 0, 3);
    __builtin_prefetch(Bp, 0, 3);

    v8f acc = {};
#pragma unroll 4
    for (int k = 0; k < K; k += 4) {
        v2f a, b;
        a.x = Ap[0];
        a.y = Ap[1];
        b.x = Bp[0];
        b.y = Bp[Nn];
        acc = __builtin_amdgcn_wmma_f32_16x16x4_f32(false, a, false, b,
                                                    (short)0, acc, false, false);
        Ap += 4;
        Bp += (size_t)4 * Nn;
    }
    const float bv = (flags & 2) ? bias[tn + lm] : 0.0f;
#pragma unroll
    for (int v = 0; v < 8; ++v) {
        float* p = C + ((size_t)(tm + 8 * h + v) * Nn + (tn + lm));
        float val = acc[v] + bv;
        if (flags & 1) val += *p;
        *p = val;
    }
}

// ---------------- TDM-staged f32 WMMA GEMM --------------------------------
// Block = 4 waves -> 16(M) x 64(N) strip.  K in 32-row chunks, double-
// buffered in LDS via TENSOR_LOAD_TO_LDS; wave0 issues + s_wait_tensorcnt,
// workgroup barriers publish the panels.  Requires M%16==0, Nn%64==0, K%32==0.
static constexpr int KC  = 32;   // K chunk
static constexpr int BNW = 64;   // N columns per block
__global__ __launch_bounds__(128) void gemm_f32_wmma_tdm(
    const float* __restrict__ A, const float* __restrict__ B,
    float* __restrict__ C, const float* __restrict__ bias,
    int M, int Nn, int K, int flags)
{
    __shared__ float smem[2*KC*BNW + 2*16*KC];   // B dbl-buf (16KB) + A dbl-buf (4KB)
    (void)M;
    const int lane = threadIdx.x & 31;
    const int wave = threadIdx.x >> 5;
    const int h    = lane >> 4;
    const int lm   = lane & 15;
    const int stripsX = Nn / BNW;
    const int tm  = (blockIdx.x / stripsX) * 16;
    const int tn0 = (blockIdx.x % stripsX) * BNW;
    const int tn  = tn0 + wave * 16;

    const unsigned ldsBase = (unsigned)(uintptr_t)(void*)smem;  // addr[31:0] == LDS offset
    float* smB = smem;                  // [buf][KC][BNW]
    float* smA = smem + 2*KC*BNW;       // [buf][16][KC]

    const int nChunks = K / KC;
    if (wave == 0) {
        tdm_load_2d(ldsBase,                      B + (size_t)0*Nn + tn0, BNW, KC, Nn);
        tdm_load_2d(ldsBase + 2u*KC*BNW*4u,       A + (size_t)tm*K,       KC, 16, K);
    }
    v8f acc = {};
    for (int c = 0; c < nChunks; ++c) {
        const int buf = c & 1;
        if (wave == 0) __builtin_amdgcn_s_wait_tensorcnt(0);   // chunk c landed
        __syncthreads();
        if (wave == 0 && c + 1 < nChunks) {                    // prefetch chunk c+1
            const int kn = (c + 1) * KC;
            const unsigned ob = (unsigned)(buf ^ 1);
            tdm_load_2d(ldsBase + ob*KC*BNW*4u,                 B + (size_t)kn*Nn + tn0, BNW, KC, Nn);
            tdm_load_2d(ldsBase + 2u*KC*BNW*4u + ob*16u*KC*4u,  A + (size_t)tm*K + kn,   KC, 16, K);
        }
        const float* bB = smB + buf * (KC * BNW);
        const float* bA = smA + buf * (16 * KC);
#pragma unroll
        for (int kk = 0; kk < KC; kk += 4) {
            const int kr = kk + 2 * h;
            v2f a, b;
            a.x = bA[lm * KC + kr];
            a.y = bA[lm * KC + kr + 1];
            b.x = bB[kr * BNW + wave * 16 + lm];
            b.y = bB[(kr + 1) * BNW + wave * 16 + lm];
            acc = __builtin_amdgcn_wmma_f32_16x16x4_f32(false, a, false, b,
                                                        (short)0, acc, false, false);
        }
        __syncthreads();   // reads done before chunk c+2 overwrites this buffer
    }
    const float bv = (flags & 2) ? bias[tn + lm] : 0.0f;
#pragma unroll
    for (int v = 0; v < 8; ++v) {
        float* p = C + ((size_t)(tm + 8 * h + v) * Nn + (tn + lm));
        float val = acc[v] + bv;
        if (flags & 1) val += *p;
        *p = val;
    }
}

// ---------------- FRFT elementwise kernels --------------------------------
__global__ void phase1_k(const float* __restrict__ c, float* __restrict__ out,
                         const float* __restrict__ idxv, const float* __restrict__ aP)
{
    size_t t = (size_t)blockIdx.x * blockDim.x + threadIdx.x;
    if (t >= (size_t)BB * NN) return;
    int k = (int)(t % NN);
    float th = HPconst * aP[0] * idxv[k];
    float sv, cv; __sincosf(th, &sv, &cv);
    float v = c[t];
    out[t]                 =  cv * v;
    out[t + (size_t)BB*NN] = -sv * v;
}
__global__ void filt_k(float* __restrict__ xa, const float* __restrict__ gf)
{
    size_t t = (size_t)blockIdx.x * blockDim.x + threadIdx.x;
    if (t >= (size_t)2 * BB * NN) return;
    xa[t] *= gf[t % NN];
}
__global__ void phase2_k(float* __restrict__ c2, const float* __restrict__ idxv,
                         const float* __restrict__ aP)
{
    size_t t = (size_t)blockIdx.x * blockDim.x + threadIdx.x;
    if (t >= (size_t)BB * NN) return;
    int k = (int)(t % NN);
    float th = HPconst * aP[0] * idxv[k];
    float sv, cv; __sincosf(th, &sv, &cv);
    float re = c2[t], im = c2[t + (size_t)BB*NN];
    c2[t]                 = re * cv - im * sv;
    c2[t + (size_t)BB*NN] = re * sv + im * cv;
}
__global__ void absY_k(const float* __restrict__ ya, float* __restrict__ Y)
{
    size_t t = (size_t)blockIdx.x * blockDim.x + threadIdx.x;
    if (t >= (size_t)BB * NN) return;
    float re = ya[t], im = ya[t + (size_t)BB*NN];
    Y[t] = sqrtf(re * re + im * im);
}

// ---------------- edge weight kernels -------------------------------------
__global__ void edge_w_k(const float* __restrict__ Y, const int* __restrict__ ei,
                         float* __restrict__ w, float* __restrict__ deg)
{
    size_t t = (size_t)blockIdx.x * blockDim.x + threadIdx.x;
    if (t >= (size_t)BB * EEn) return;
    int g = (int)(t / EEn), e = (int)(t % EEn);
    const int* eb = ei + (size_t)g * 2 * EEn;
    int s = eb[e], d = eb[EEn + e];
    float yu = Y[(size_t)g*NN + s], yv = Y[(size_t)g*NN + d];
    float dd = fabsf(yu - yv) / (fabsf(yu) + fabsf(yv) + 1e-8f);
    float wv = fmaxf(1.0f - dd, 1e-6f);
    w[t] = wv;
    atomicAdd(&deg[(size_t)g*NN + s], wv);
}
__global__ void edge_wn_k(const int* __restrict__ ei, const float* __restrict__ w,
                          const float* __restrict__ deg, float* __restrict__ wn)
{
    size_t t = (size_t)blockIdx.x * blockDim.x + threadIdx.x;
    if (t >= (size_t)BB * EEn) return;
    int g = (int)(t / EEn), e = (int)(t % EEn);
    const int* eb = ei + (size_t)g * 2 * EEn;
    int s = eb[e], d = eb[EEn + e];
    float du = deg[(size_t)g*NN + s], dv = deg[(size_t)g*NN + d];
    float iu = du > 0.0f ? rsqrtf(du) : 0.0f;
    float iv = dv > 0.0f ? rsqrtf(dv) : 0.0f;
    wn[t] = iu * w[t] * iv;
}

// ---------------- Chebyshev propagation -----------------------------------
__global__ void prop_scalar_k(const int* __restrict__ ei, const float* __restrict__ wn,
                              const float* __restrict__ tin, float* __restrict__ tout)
{
    size_t t = (size_t)blockIdx.x * blockDim.x + threadIdx.x;
    if (t >= (size_t)BB * EEn) return;
    int g = (int)(t / EEn), e = (int)(t % EEn);
    const int* eb = ei + (size_t)g * 2 * EEn;
    int s = eb[e], d = eb[EEn + e];
    atomicAdd(&tout[(size_t)g*NN + d], -wn[t] * tin[(size_t)g*NN + s]);
}
static constexpr int EPB = 64;  // edges per block (vector prop)
__global__ __launch_bounds__(HIDc) void prop_vec_k(
    const int* __restrict__ ei, const float* __restrict__ wn,
    const float* __restrict__ Tin, float* __restrict__ Tout)
{
    const int blocksPerG = EEn / EPB;
    int g  = blockIdx.x / blocksPerG;
    int e0 = (blockIdx.x % blocksPerG) * EPB;
    int j  = threadIdx.x;
    const int* eb = ei + (size_t)g * 2 * EEn;
    for (int t = 0; t < EPB; ++t) {
        int e = e0 + t;
        int s = eb[e], d = eb[EEn + e];
        float w = -wn[(size_t)g*EEn + e];
        atomicAdd(&Tout[((size_t)g*NN + d) * HIDc + j],
                  w * Tin[((size_t)g*NN + s) * HIDc + j]);
    }
}
__global__ void finalize2_k(float* __restrict__ t, const float* __restrict__ t0, size_t n)
{
    size_t i = (size_t)blockIdx.x * blockDim.x + threadIdx.x;
    if (i >= n) return;
    t[i] = 2.0f * t[i] - t0[i];
}

// ---------------- layer 1 (W1 is (3,1,HID): outer product) ----------------
__global__ void layer1_k(const float* __restrict__ x, const float* __restrict__ t1,
                         const float* __restrict__ t2, const float* __restrict__ W1,
                         const float* __restrict__ b1, float* __restrict__ H)
{
    size_t t = (size_t)blockIdx.x * blockDim.x + threadIdx.x;
    if (t >= (size_t)BB * NN * HIDc) return;
    int j = (int)(t % HIDc);
    size_t gn = t / HIDc;
    H[t] = x[gn]*W1[j] + t1[gn]*W1[HIDc + j] + t2[gn]*W1[2*HIDc + j] + b1[j];
}

// ---------------- graph norm + relu ---------------------------------------
__global__ __launch_bounds__(HIDc) void norm_relu_k(
    float* __restrict__ H, const float* __restrict__ al,
    const float* __restrict__ ga, const float* __restrict__ be)
{
    int g = blockIdx.x, j = threadIdx.x;
    float* base = H + (size_t)g * NN * HIDc + j;
    float sum = 0.0f, sq = 0.0f;
    for (int n = 0; n < NN; ++n) { float v = base[(size_t)n*HIDc]; sum += v; sq += v*v; }
    float mu = sum / NN;
    float A  = al[j];
    float var = sq / NN - (2.0f*A - A*A) * mu * mu;
    float rs  = rsqrtf(var + 1e-5f);
    float G = ga[j], Bv = be[j];
    for (int n = 0; n < NN; ++n) {
        float v = base[(size_t)n*HIDc];
        v = G * (v - A*mu) * rs + Bv;
        base[(size_t)n*HIDc] = fmaxf(v, 0.0f);
    }
}

// ---------------- max-pool over nodes + classifier ------------------------
__global__ __launch_bounds__(HIDc) void final_k(
    const float* __restrict__ H, const float* __restrict__ Wl,
    const float* __restrict__ bl, float* __restrict__ out)
{
    __shared__ float gm[HIDc];
    int g = blockIdx.x, j = threadIdx.x;
    const float* base = H + (size_t)g * NN * HIDc + j;
    float m = -INFINITY;
    for (int n = 0; n < NN; ++n) m = fmaxf(m, base[(size_t)n*HIDc]);
    gm[j] = m;
    __syncthreads();
    if (j < NCLSc) {
        float s = bl[j];
        for (int k = 0; k < HIDc; ++k) s += gm[k] * Wl[k*NCLSc + j];
        out[g*NCLSc + j] = s;
    }
}

// ======================= host: DFRFT basis (constant) =====================
static void jacobi_eigh(std::vector<double>& A, std::vector<double>& V, int n)
{
    for (size_t i = 0; i < (size_t)n*n; ++i) V[i] = 0.0;
    for (int i = 0; i < n; ++i) V[(size_t)i*n + i] = 1.0;
    for (int sw = 0; sw < 24; ++sw) {
        double off = 0.0;
        for (int p = 0; p < n; ++p)
            for (int q = p+1; q < n; ++q) off = std::max(off, std::fabs(A[(size_t)p*n+q]));
        if (off < 1e-11) break;
        for (int p = 0; p < n-1; ++p)
            for (int q = p+1; q < n; ++q) {
                double apq = A[(size_t)p*n+q];
                if (std::fabs(apq) < 1e-13) continue;
                double app = A[(size_t)p*n+p], aqq = A[(size_t)q*n+q];
                double tau = (aqq - app) / (2.0*apq);
                double tt  = (tau >= 0.0 ? 1.0 : -1.0) / (std::fabs(tau) + std::sqrt(1.0 + tau*tau));
                double c = 1.0 / std::sqrt(1.0 + tt*tt);
                double s = tt * c;
                double* Rp = &A[(size_t)p*n]; double* Rq = &A[(size_t)q*n];
                for (int i = 0; i < n; ++i) {
                    double ap = Rp[i], aq = Rq[i];
                    Rp[i] = c*ap - s*aq; Rq[i] = s*ap + c*aq;
                }
                for (int i = 0; i < n; ++i) {
                    double ap = A[(size_t)i*n+p], aq = A[(size_t)i*n+q];
                    A[(size_t)i*n+p] = c*ap - s*aq; A[(size_t)i*n+q] = s*ap + c*aq;
                }
                for (int i = 0; i < n; ++i) {
                    double vp = V[(size_t)i*n+p], vq = V[(size_t)i*n+q];
                    V[(size_t)i*n+p] = c*vp - s*vq; V[(size_t)i*n+q] = s*vp + c*vq;
                }
            }
    }
}

static void build_basis(float* Ef, float* ETf, float* idxf, float* gff)
{
    const int n = NN, half = n/2 + 1;             // 513
    const double r = 1.0 / std::sqrt(2.0);
    std::vector<int> prN(n,0), pcN(n,0);
    std::vector<int> prI(n*2), pcI(n*2);
    std::vector<double> prV(n*2), pcV(n*2);
    auto addP = [&](int i, int j, double v){
        prI[i*2+prN[i]] = j; prV[i*2+prN[i]] = v; prN[i]++;
        pcI[j*2+pcN[j]] = i; pcV[j*2+pcN[j]] = v; pcN[j]++;
    };
    addP(0,0,1.0); addP(n/2, n/2, 1.0);
    for (int k = 1; k < (n+1)/2; ++k) {
        addP(k, k, r); addP(k, n-k, r); addP(n-k, k, r); addP(n-k, n-k, -r);
    }
    std::vector<double> T((size_t)n*n, 0.0);
    for (int i = 0; i < n; ++i)
        for (int e = 0; e < prN[i]; ++e) {
            int k = prI[i*2+e]; double v = prV[i*2+e];
            T[(size_t)i*n + k]           += v * 2.0 * std::cos(2.0*M_PI*k/n);
            T[(size_t)i*n + (k+1)%n]     += v;
            T[(size_t)i*n + (k-1+n)%n]   += v;
        }
    std::vector<double> CS((size_t)n*n, 0.0);
    for (int i = 0; i < n; ++i)
        for (int j = 0; j < n; ++j) {
            double s = 0.0;
            for (int e = 0; e < prN[j]; ++e) s += T[(size_t)i*n + prI[j*2+e]] * prV[j*2+e];
            CS[(size_t)i*n + j] = s;
        }
    const int ne = half, no = n - half;           // 513, 511
    std::vector<double> Ev((size_t)ne*ne), Vev((size_t)ne*ne);
    std::vector<double> Od((size_t)no*no), Vod((size_t)no*no);
    for (int i = 0; i < ne; ++i) for (int j = 0; j < ne; ++j) Ev[(size_t)i*ne+j] = CS[(size_t)i*n+j];
    for (int i = 0; i < no; ++i) for (int j = 0; j < no; ++j) Od[(size_t)i*no+j] = CS[(size_t)(i+half)*n + (j+half)];
    jacobi_eigh(Ev, Vev, ne);
    jacobi_eigh(Od, Vod, no);
    std::vector<int> se(ne), so(no);
    for (int i = 0; i < ne; ++i) se[i] = i;
    for (int i = 0; i < no; ++i) so[i] = i;
    std::stable_sort(se.begin(), se.end(), [&](int a, int b){ return Ev[(size_t)a*ne+a] > Ev[(size_t)b*ne+b]; });
    std::stable_sort(so.begin(), so.end(), [&](int a, int b){ return Od[(size_t)a*no+a] > Od[(size_t)b*no+b]; });
    auto EeVal = [&](int i, int k)->double {
        double s = 0.0;
        for (int e = 0; e < pcN[i]; ++e) {
            int j = pcI[i*2+e];
            if (j < half) s += pcV[i*2+e] * Vev[(size_t)j*ne + se[k]];
        }
        return s;
    };
    auto EoVal = [&](int i, int k)->double {
        double s = 0.0;
        for (int e = 0; e < pcN[i]; ++e) {
            int j = pcI[i*2+e];
            if (j >= half) s += pcV[i*2+e] * Vod[(size_t)(j-half)*no + so[k]];
        }
        return s;
    };
    for (int i = 0; i < n; ++i) {
        for (int k = 0; k < no; ++k) {
            float ve = (float)EeVal(i, k), vo = (float)EoVal(i, k);
            Ef[(size_t)i*n + 2*k]     = ve;  ETf[(size_t)(2*k)*n + i]   = ve;
            Ef[(size_t)i*n + 2*k + 1] = vo;  ETf[(size_t)(2*k+1)*n + i] = vo;
        }
        for (int k = no; k < ne; ++k) {
            float ve = (float)EeVal(i, k);
            int col = no + k;                     // 1022, 1023
            Ef[(size_t)i*n + col] = ve;  ETf[(size_t)col*n + i] = ve;
        }
    }
    for (int k = 0; k < n-1; ++k) idxf[k] = (float)k;
    idxf[n-1] = (float)n;
    int cut = 154;                                 // round(0.15*1024)
    for (int k = 0; k < n; ++k) gff[k] = 0.0f;
    for (int k = 0; k < cut; ++k) { gff[k] = 1.0f; gff[n-1-k] = 1.0f; }
}

// ============================== launch ====================================
static inline int ceil_div(size_t a, int b) { return (int)((a + b - 1) / b); }

extern "C" void kernel_launch(void* const* d_in, const int* in_sizes, int n_in,
                              void* d_out, int out_size, void* d_ws, size_t ws_size,
                              hipStream_t stream)
{
    (void)in_sizes; (void)n_in; (void)out_size; (void)d_ws; (void)ws_size;
    const float* x   = (const float*)d_in[0];
    const int*   ei  = (const int*)  d_in[1];
    const float* aP  = (const float*)d_in[2];
    const float* W1  = (const float*)d_in[3];
    const float* b1  = (const float*)d_in[4];
    const float* al1 = (const float*)d_in[5];
    const float* ga1 = (const float*)d_in[6];
    const float* be1 = (const float*)d_in[7];
    const float* W2  = (const float*)d_in[8];
    const float* b2  = (const float*)d_in[9];
    const float* al2 = (const float*)d_in[10];
    const float* ga2 = (const float*)d_in[11];
    const float* be2 = (const float*)d_in[12];
    const float* W3  = (const float*)d_in[13];
    const float* b3  = (const float*)d_in[14];
    const float* al3 = (const float*)d_in[15];
    const float* ga3 = (const float*)d_in[16];
    const float* be3 = (const float*)d_in[17];
    const float* Wl  = (const float*)d_in[18];
    const float* bl  = (const float*)d_in[19];
    float* out = (float*)d_out;

    // Constant DFRFT basis: recomputed each call into persistent host statics.
    static float hE[(size_t)NN*NN], hET[(size_t)NN*NN], hIdx[NN], hGf[NN];
    build_basis(hE, hET, hIdx, hGf);

    float* heap = nullptr;
    (void)hipGetSymbolAddress((void**)&heap, HIP_SYMBOL(g_heap));
    float* E   = heap + OFF_E;
    float* ET  = heap + OFF_ET;
    float* IDX = heap + OFF_IDX;
    float* GF  = heap + OFF_GF;
    float* BA  = heap + OFF_BA;
    float* BBf = heap + OFF_BBUF;
    float* Y   = heap + OFF_Y;
    float* Wb  = heap + OFF_W;
    float* WN  = heap + OFF_WN;
    float* DEG = heap + OFF_DEG;
    float* T1S = heap + OFF_T1S;
    float* T2S = heap + OFF_T2S;
    float* HA  = heap + OFF_HA;
    float* HB  = heap + OFF_HB;
    float* HC  = heap + OFF_HC;
    float* HD  = heap + OFF_HD;

    (void)hipMemcpyAsync(E,   hE,   sizeof(float)*(size_t)NN*NN, hipMemcpyHostToDevice, stream);
    (void)hipMemcpyAsync(ET,  hET,  sizeof(float)*(size_t)NN*NN, hipMemcpyHostToDevice, stream);
    (void)hipMemcpyAsync(IDX, hIdx, sizeof(float)*NN,            hipMemcpyHostToDevice, stream);
    (void)hipMemcpyAsync(GF,  hGf,  sizeof(float)*NN,            hipMemcpyHostToDevice, stream);

    // ---------------- FRFT ----------------
    {
        int tiles1 = (BB/16) * (NN/16);
        gemm_f32_wmma<<<ceil_div(tiles1,4),128,0,stream>>>(x, E, BBf, nullptr, BB, NN, NN, 0);
        phase1_k<<<ceil_div((size_t)BB*NN,256),256,0,stream>>>(BBf, BA, IDX, aP);
        int blocks2 = (2*BB/16) * (NN/BNW);        // 16 x 16 = 256 blocks
        gemm_f32_wmma_tdm<<<blocks2,128,0,stream>>>(BA, ET, BBf, nullptr, 2*BB, NN, NN, 0);
        filt_k<<<ceil_div((size_t)2*BB*NN,256),256,0,stream>>>(BBf, GF);
        gemm_f32_wmma_tdm<<<blocks2,128,0,stream>>>(BBf, E, BA, nullptr, 2*BB, NN, NN, 0);
        phase2_k<<<ceil_div((size_t)BB*NN,256),256,0,stream>>>(BA, IDX, aP);
        gemm_f32_wmma_tdm<<<blocks2,128,0,stream>>>(BA, ET, BBf, nullptr, 2*BB, NN, NN, 0);
        absY_k<<<ceil_div((size_t)BB*NN,256),256,0,stream>>>(BBf, Y);
    }

    // ---------------- edge weights ----------------
    (void)hipMemsetAsync(DEG, 0, sizeof(float)*(size_t)BB*NN, stream);
    edge_w_k <<<ceil_div((size_t)BB*EEn,256),256,0,stream>>>(Y, ei, Wb, DEG);
    edge_wn_k<<<ceil_div((size_t)BB*EEn,256),256,0,stream>>>(ei, Wb, DEG, WN);

    // ---------------- layer 1 (scalar Chebyshev) ----------------
    (void)hipMemsetAsync(T1S, 0, sizeof(float)*(size_t)BB*NN, stream);
    (void)hipMemsetAsync(T2S, 0, sizeof(float)*(size_t)BB*NN, stream);
    prop_scalar_k<<<ceil_div((size_t)BB*EEn,256),256,0,stream>>>(ei, WN, x,   T1S);
    prop_scalar_k<<<ceil_div((size_t)BB*EEn,256),256,0,stream>>>(ei, WN, T1S, T2S);
    finalize2_k<<<ceil_div((size_t)BB*NN,256),256,0,stream>>>(T2S, x, (size_t)BB*NN);
    layer1_k<<<ceil_div((size_t)BB*NN*HIDc,256),256,0,stream>>>(x, T1S, T2S, W1, b1, HA);
    norm_relu_k<<<BB,HIDc,0,stream>>>(HA, al1, ga1, be1);

    const int chebBlocks = ((BB*NN)/16) * (HIDc/BNW);   // 8192 * 2
    const int propGrid   = BB * (EEn / EPB);
    const size_t Hn      = (size_t)BB*NN*HIDc;

    // ---------------- layer 2 ----------------
    (void)hipMemsetAsync(HB, 0, sizeof(float)*Hn, stream);
    (void)hipMemsetAsync(HC, 0, sizeof(float)*Hn, stream);
    prop_vec_k<<<propGrid,HIDc,0,stream>>>(ei, WN, HA, HB);
    prop_vec_k<<<propGrid,HIDc,0,stream>>>(ei, WN, HB, HC);
    finalize2_k<<<ceil_div(Hn,256),256,0,stream>>>(HC, HA, Hn);
    gemm_f32_wmma_tdm<<<chebBlocks,128,0,stream>>>(HA, W2,              HD, b2,      BB*NN, HIDc, HIDc, 2);
    gemm_f32_wmma_tdm<<<chebBlocks,128,0,stream>>>(HB, W2 + HIDc*HIDc,  HD, nullptr, BB*NN, HIDc, HIDc, 1);
    gemm_f32_wmma_tdm<<<chebBlocks,128,0,stream>>>(HC, W2 + 2*HIDc*HIDc,HD, nullptr, BB*NN, HIDc, HIDc, 1);
    norm_relu_k<<<BB,HIDc,0,stream>>>(HD, al2, ga2, be2);

    // ---------------- layer 3 ----------------
    (void)hipMemsetAsync(HA, 0, sizeof(float)*Hn, stream);
    (void)hipMemsetAsync(HB, 0, sizeof(float)*Hn, stream);
    prop_vec_k<<<propGrid,HIDc,0,stream>>>(ei, WN, HD, HA);
    prop_vec_k<<<propGrid,HIDc,0,stream>>>(ei, WN, HA, HB);
    finalize2_k<<<ceil_div(Hn,256),256,0,stream>>>(HB, HD, Hn);
    gemm_f32_wmma_tdm<<<chebBlocks,128,0,stream>>>(HD, W3,              HC, b3,      BB*NN, HIDc, HIDc, 2);
    gemm_f32_wmma_tdm<<<chebBlocks,128,0,stream>>>(HA, W3 + HIDc*HIDc,  HC, nullptr, BB*NN, HIDc, HIDc, 1);
    gemm_f32_wmma_tdm<<<chebBlocks,128,0,stream>>>(HB, W3 + 2*HIDc*HIDc,HC, nullptr, BB*NN, HIDc, HIDc, 1);
    norm_relu_k<<<BB,HIDc,0,stream>>>(HC, al3, ga3, be3);

    // ---------------- pool + classifier ----------------
    final_k<<<BB,HIDc,0,stream>>>(HC, Wl, bl, out);
}